// MappingRotate_47974784696539
// MI455X (gfx1250) — compile-verified
//
#include <hip/hip_runtime.h>
#include <stdint.h>

// Problem constants (fixed by the reference).
#define O_  512
#define I_  256
#define E_  72                 // Or * kH * kW
#define R_  8
#define IC  64                 // i-rows staged per block
#define NCHUNK (I_ / IC)       // 4
#define TPB 288                // 9 wave32; 288 = 16 rows x 18 float4-columns
#define ROW_F4  (E_ / 4)       // 18 float4 per row
#define TILE_F4 ((IC * E_) / 4) // 1152 = 4 * TPB  (uniform async-stage loop)
#define ROWS_PER_PASS (TPB / ROW_F4)  // 16
#define NPASS (IC / ROWS_PER_PASS)    // 4

typedef float v4f __attribute__((ext_vector_type(4)));
typedef int   v4i __attribute__((ext_vector_type(4)));

// out[((o*R + r)*I + i)*E + j] = in[(o*I + i)*E + inv[r][j]]
// where inv[r][ idx[e*R + r] - 1 ] = e.
__global__ __launch_bounds__(TPB) void mapping_rotate_kernel(
    const float* __restrict__ in,
    const int*   __restrict__ idx,
    float*       __restrict__ out)
{
    __shared__ alignas(16) float s_tile[IC * E_];      // 18432 B
    __shared__ alignas(16) int   s_inv[R_ * E_];       // 2304 B

    const int t  = threadIdx.x;
    const int b  = blockIdx.x;
    const int o  = b / NCHUNK;
    const int i0 = (b - o * NCHUNK) * IC;

    // ---- Build inverse permutations in LDS (scatter; perms -> no conflicts) ----
    if (t < E_) {
        #pragma unroll
        for (int r = 0; r < R_; ++r) {
            int tg = idx[t * R_ + r] - 1;      // tgt[r][t] in [0, E)
            s_inv[r * E_ + tg] = t;
        }
    }

    // ---- Async-stage the input tile (64 rows x 72 floats) into LDS ----
    // Global byte offset (o*I + i0)*E_*4 is a multiple of 18432 -> 16B aligned.
    // 1152 float4 / 288 threads = exactly 4 uniform iterations (no divergent tail).
    const float* gsrc = in + ((size_t)o * I_ + (size_t)i0) * E_;
    const uint32_t tb = (uint32_t)(uintptr_t)s_tile;   // low 32 bits of flat addr = LDS offset
    #pragma unroll
    for (int p = 0; p < TILE_F4 / TPB; ++p) {
        int k = t + p * TPB;
        uint32_t ldsoff = tb + (uint32_t)k * 16u;
        const float* g  = gsrc + (size_t)k * 4;
        asm volatile("global_load_async_to_lds_b128 %0, %1, off"
                     :: "v"(ldsoff), "v"(g)
                     : "memory");
    }
    asm volatile("s_wait_asynccnt 0" ::: "memory");
    __syncthreads();

    // ---- Fixed (j4, row-lane) mapping: the division happens once, not per iter ----
    const int j4  = t % ROW_F4;        // float4 column within a 72-float row
    const int ii0 = t / ROW_F4;        // 0..15: row slot within a pass

    #pragma unroll 1
    for (int r = 0; r < R_; ++r) {
        // Hoisted: the 4 permutation indices this thread uses for all rows of r.
        v4i iv = *(const v4i*)(s_inv + r * E_ + j4 * 4);   // 16B-aligned LDS load
        float* outr = out + (((size_t)o * R_ + (size_t)r) * I_ + (size_t)i0) * E_
                          + (size_t)j4 * 4;
        #pragma unroll
        for (int p = 0; p < NPASS; ++p) {
            int ii = ii0 + p * ROWS_PER_PASS;
            const float* row = s_tile + ii * E_;
            v4f v;
            v.x = row[iv.x];
            v.y = row[iv.y];
            v.z = row[iv.z];
            v.w = row[iv.w];
            // Streamed output: non-temporal so the 8x-reused input stays hot in L2.
            __builtin_nontemporal_store(v, (v4f*)(outr + (size_t)ii * E_));
        }
    }
}

extern "C" void kernel_launch(void* const* d_in, const int* in_sizes, int n_in,
                              void* d_out, int out_size, void* d_ws, size_t ws_size,
                              hipStream_t stream) {
    const float* in  = (const float*)d_in[0];   // (512, 256, 8, 3, 3) fp32
    const int*   idx = (const int*)d_in[1];     // (8, 9, 8) int32
    float*       out = (float*)d_out;           // (4096, 256, 8, 3, 3) fp32

    dim3 grid(O_ * NCHUNK);   // 2048 blocks
    dim3 block(TPB);          // 288 threads = 9 wave32
    mapping_rotate_kernel<<<grid, block, 0, stream>>>(in, idx, out);
}